// HeteroGraphSAGE_26482768347335
// MI455X (gfx1250) — compile-verified
//
#include <hip/hip_runtime.h>

#define CCH 128
#define WPAD 130                      // padded LDS row stride (words) to spread banks
#define GEMM_WAVES 8                  // 8 waves x 16 rows = 128 rows per block
#define SMEM_BYTES (2 * CCH * WPAD * 4)

typedef float v2f __attribute__((ext_vector_type(2)));
typedef float v8f __attribute__((ext_vector_type(8)));

// ---------------- zero fill (workspace is poisoned; re-zero every call) ---------------
__global__ __launch_bounds__(256) void hsage_zero_kernel(float* __restrict__ p, long n) {
    long i = (long)blockIdx.x * 256 + threadIdx.x;
    if (i < n) p[i] = 0.0f;
}

// ---------------- edge scatter: summed[dst] += x_src[src]; cnt[dst] += 1 --------------
// One wave32 per edge; each lane moves 16 bytes (float4) -> fully coalesced 512B gather
// per edge, then 4 native f32 atomics into the (L2-resident) accumulation buffer.
__global__ __launch_bounds__(256) void hsage_scatter_kernel(
    const float* __restrict__ xsrc, const int* __restrict__ src,
    const int* __restrict__ dst, float* __restrict__ summed,
    float* __restrict__ cnt, int E)
{
    int tid = blockIdx.x * 256 + threadIdx.x;
    int e = tid >> 5;
    if (e >= E) return;
    int lane = tid & 31;
    int s = src[e];
    int d = dst[e];
    float4 v = ((const float4*)(xsrc + (size_t)s * CCH))[lane];
    float* sd = summed + (size_t)d * CCH + lane * 4;
    unsafeAtomicAdd(sd + 0, v.x);
    unsafeAtomicAdd(sd + 1, v.y);
    unsafeAtomicAdd(sd + 2, v.z);
    unsafeAtomicAdd(sd + 3, v.w);
    if (lane == 0) unsafeAtomicAdd(cnt + d, 1.0f);
}

// -------- fused: out = relu(LN(mean @ Wl^T + bl + x_dst @ Wr^T)) via V_WMMA_F32_16X16X4_F32
// 8 waves/block; each wave owns a 16-row node block and the full 16x128 output tile
// (8 N-tiles x v8f accumulators). Both weight matrices are staged once per block into
// LDS (padded rows), so inner-loop B fragments come from ds_load_b64 instead of L2.
// A layout  (16x4 f32): lane row = lane&15, K = 2*(lane>>4) + vgpr     (ISA 7.12.2)
// B layout  (4x16 f32): lane col = lane&15, K = 2*(lane>>4) + vgpr
// C/D layout(16x16 f32): row = vgpr + 8*(lane>>4), col = lane&15
__global__ __launch_bounds__(256) void hsage_gemm_ln_relu_kernel(
    const float* __restrict__ summed, const float* __restrict__ cnt,
    const float* xdst,                       // may alias `out` (in-place, wave-local rows)
    const float* __restrict__ Wl, const float* __restrict__ bl,
    const float* __restrict__ Wr,
    const float* __restrict__ lnw, const float* __restrict__ lnb,
    float* out, int M)
{
    extern __shared__ float smem[];
    float* sWl = smem;                       // [CCH][WPAD]
    float* sWr = smem + CCH * WPAD;          // [CCH][WPAD]

    // ---- cooperative weight stage: global-coalesced reads, conflict-free LDS writes ----
    for (int i = threadIdx.x; i < CCH * CCH; i += 256) {
        const int n = i >> 7;
        const int k = i & (CCH - 1);
        sWl[n * WPAD + k] = Wl[i];
        sWr[n * WPAD + k] = Wr[i];
    }
    __syncthreads();

    const int lane = threadIdx.x & 31;
    const int wave = threadIdx.x >> 5;
    const int m0   = (blockIdx.x * GEMM_WAVES + wave) * 16;
    if (m0 < M) {                            // wave-uniform: EXEC all-ones inside
        const int half = lane >> 4;          // 0 or 1
        const int l15  = lane & 15;
        const int arow = m0 + l15;           // A-matrix row owned by this lane
        const int koff = half * 2;           // K offset within each group of 4

        const float scale = 1.0f / fmaxf(cnt[arow], 1.0f);  // mean = summed / max(cnt,1)

        const float2* __restrict__ aSum = (const float2*)(summed + (size_t)arow * CCH);
        const float2*              aX   = (const float2*)(xdst   + (size_t)arow * CCH);

        v8f acc[8] = {};                     // 8 N-tiles -> 16x128 outputs per wave

        for (int kk = 0; kk < CCH; kk += 4) {
            const int k0 = kk + koff;
            const int kh = k0 >> 1;          // float2 index (8B-aligned loads)
            float2 s2 = aSum[kh];
            float2 x2 = aX[kh];
            v2f a1; a1.x = s2.x * scale; a1.y = s2.y * scale;  // mean path (scale fused)
            v2f a2; a2.x = x2.x;         a2.y = x2.y;          // self path
#pragma unroll
            for (int t = 0; t < 8; ++t) {
                const int ncol = t * 16 + l15;
                // B[k][n] = W[n][k]  (out = A @ W^T); padded-row LDS, 8B-aligned
                float2 w1 = *(const float2*)(sWl + ncol * WPAD + k0);
                float2 w2 = *(const float2*)(sWr + ncol * WPAD + k0);
                v2f b1; b1.x = w1.x; b1.y = w1.y;
                v2f b2; b2.x = w2.x; b2.y = w2.y;
                acc[t] = __builtin_amdgcn_wmma_f32_16x16x4_f32(
                    false, a1, false, b1, (short)0, acc[t], false, false);
                acc[t] = __builtin_amdgcn_wmma_f32_16x16x4_f32(
                    false, a2, false, b2, (short)0, acc[t], false, false);
            }
        }

        // ---- epilogue: + bias, LayerNorm over 128 outputs of each row, ReLU, store ----
        float p[8], q[8];
#pragma unroll
        for (int r = 0; r < 8; ++r) { p[r] = 0.0f; q[r] = 0.0f; }
#pragma unroll
        for (int t = 0; t < 8; ++t) {
            const int ncol = t * 16 + l15;
            const float b = bl[ncol];
#pragma unroll
            for (int r = 0; r < 8; ++r) {
                float v = acc[t][r] + b;
                acc[t][r] = v;
                p[r] += v;
                q[r] += v * v;
            }
        }
        // Row r (+8 for upper half) lives across the 16 lanes of this wave-half:
        // xor masks 1,2,4,8 keep the butterfly inside each half.
#pragma unroll
        for (int m = 1; m < 16; m <<= 1) {
#pragma unroll
            for (int r = 0; r < 8; ++r) {
                p[r] += __shfl_xor(p[r], m, 32);
                q[r] += __shfl_xor(q[r], m, 32);
            }
        }
#pragma unroll
        for (int t = 0; t < 8; ++t) {
            const int ncol = t * 16 + l15;
            const float w  = lnw[ncol];
            const float bb = lnb[ncol];
#pragma unroll
            for (int r = 0; r < 8; ++r) {
                const float mu   = p[r] * (1.0f / CCH);
                const float var  = q[r] * (1.0f / CCH) - mu * mu;
                const float rstd = rsqrtf(var + 1e-5f);
                float v = (acc[t][r] - mu) * rstd * w + bb;
                v = fmaxf(v, 0.0f);
                out[(size_t)(m0 + half * 8 + r) * CCH + ncol] = v;
            }
        }
    }
}

extern "C" void kernel_launch(void* const* d_in, const int* in_sizes, int n_in,
                              void* d_out, int out_size, void* d_ws, size_t ws_size,
                              hipStream_t stream) {
    (void)n_in; (void)out_size; (void)ws_size;
    const float* x_user0  = (const float*)d_in[0];
    const float* x_item0  = (const float*)d_in[1];
    const float* Wl_ut    = (const float*)d_in[2];
    const float* bl_ut    = (const float*)d_in[3];
    const float* Wr_ut    = (const float*)d_in[4];
    const float* Wl_iu    = (const float*)d_in[5];
    const float* bl_iu    = (const float*)d_in[6];
    const float* Wr_iu    = (const float*)d_in[7];
    const float* lnw_user = (const float*)d_in[8];
    const float* lnb_user = (const float*)d_in[9];
    const float* lnw_item = (const float*)d_in[10];
    const float* lnb_item = (const float*)d_in[11];
    const int*   ei_ut    = (const int*)d_in[12];
    const int*   ei_iu    = (const int*)d_in[13];

    const int N = in_sizes[0] / CCH;
    const int L = in_sizes[2] / (CCH * CCH);
    const int E = in_sizes[12] / 2;

    float* out_user = (float*)d_out;
    float* out_item = (float*)d_out + (size_t)N * CCH;

    // workspace: [summed_item | summed_user | cnt_item | cnt_user] (contiguous floats)
    float* summed_item = (float*)d_ws;
    float* summed_user = summed_item + (size_t)N * CCH;
    float* cnt_item    = summed_user + (size_t)N * CCH;
    float* cnt_user    = cnt_item + N;

    const long zn    = 2L * N * CCH + 2L * N;
    const int  zgrid = (int)((zn + 255) / 256);
    const int  sgrid = (int)(((long)E * 32 + 255) / 256);
    const int  mblk  = (N + 15) / 16;
    const int  mgrid = (mblk + GEMM_WAVES - 1) / GEMM_WAVES;

    // allow >64KB dynamic LDS (320KB per WGP on CDNA5)
    hipFuncSetAttribute((const void*)hsage_gemm_ln_relu_kernel,
                        hipFuncAttributeMaxDynamicSharedMemorySize, SMEM_BYTES);

    const float* cu = x_user0;
    const float* ci = x_item0;
    for (int l = 0; l < L; ++l) {
        hsage_zero_kernel<<<zgrid, 256, 0, stream>>>(summed_item, zn);
        // user -> item
        hsage_scatter_kernel<<<sgrid, 256, 0, stream>>>(cu, ei_ut, ei_ut + E,
                                                        summed_item, cnt_item, E);
        // item -> user
        hsage_scatter_kernel<<<sgrid, 256, 0, stream>>>(ci, ei_iu, ei_iu + E,
                                                        summed_user, cnt_user, E);
        const size_t wo = (size_t)l * CCH * CCH;
        const size_t bo = (size_t)l * CCH;
        hsage_gemm_ln_relu_kernel<<<mgrid, 256, SMEM_BYTES, stream>>>(
            summed_item, cnt_item, ci, Wl_ut + wo, bl_ut + bo, Wr_ut + wo,
            lnw_item + bo, lnb_item + bo, out_item, N);
        hsage_gemm_ln_relu_kernel<<<mgrid, 256, SMEM_BYTES, stream>>>(
            summed_user, cnt_user, cu, Wl_iu + wo, bl_iu + bo, Wr_iu + wo,
            lnw_user + bo, lnb_user + bo, out_user, N);
        cu = out_user;  // layer >=1 runs in place on d_out (wave-local rows: safe)
        ci = out_item;
    }
}